// ViterbiLoss_64879775973939
// MI455X (gfx1250) — compile-verified
//
#include <hip/hip_runtime.h>
#include <stdint.h>

// CRF forward-algorithm loss for B=128, T=256, K=50 on gfx1250.
// One block per batch row; depth-4 async global->LDS pipeline; per-step
// logsumexp factored into exp + f32 WMMA matvec (V_WMMA_F32_16X16X4_F32).

#define KTAGS 50
#define KKE   (KTAGS * KTAGS)   // 2500 floats per (b,t) tile
#define KPAD  64                // padded K for WMMA
#define EST   72                // E row stride (floats); 72*4 mod 64 banks avoids half-wave overlap
#define NTH   128               // 4 waves (wave32)
#define NBUF  4                 // async pipeline depth
#define START_TAG 48
#define END_TAG   49

typedef float v2f __attribute__((ext_vector_type(2)));
typedef float v8f __attribute__((ext_vector_type(8)));

__device__ __forceinline__ void async_ld_b32(uint32_t lds_off, const float* g) {
  // GLOBAL_LOAD_ASYNC_TO_LDS_B32, GV mode: vdst = LDS byte offset, vaddr = 64-bit address.
  asm volatile("global_load_async_to_lds_b32 %0, %1, off"
               :: "v"(lds_off), "v"(g) : "memory");
}

// Each tile issues 19 or 20 async loads per thread (2500/128). Waiting for the
// oldest outstanding tile with `ga` newer tiles in flight: threshold = 19*ga
// (safe lower bound across threads; async loads complete in order).
__device__ __forceinline__ void wait_async_groups(int ga) {
  if (ga >= 2)      asm volatile("s_wait_asynccnt 38" ::: "memory");
  else if (ga == 1) asm volatile("s_wait_asynccnt 19" ::: "memory");
  else              asm volatile("s_wait_asynccnt 0"  ::: "memory");
}

__global__ __launch_bounds__(NTH)
void crf_fwd_kernel(const float* __restrict__ scores,
                    const int*   __restrict__ targets,
                    const int*   __restrict__ lengths,
                    float*       __restrict__ partial,
                    int T) {
  __shared__ float raw[NBUF][KKE];     // streamed score tiles
  __shared__ float E[KPAD * EST];      // exp(s - colmax), zero-padded to 64x64
  __shared__ float alpha[KPAD];
  __shared__ float pvec[KPAD];         // exp(alpha - am), zero-padded
  __shared__ float cmax[KPAD];
  __shared__ float red[NTH];
  __shared__ float amax_s;
  __shared__ float vout[KPAD];

  const int tid = threadIdx.x;
  const int b   = blockIdx.x;
  const int len = lengths[b];
  const float* sb = scores + (size_t)b * T * KKE;

  // One-time zero init: pad rows/cols of E and pad entries of pvec stay 0 forever.
  for (int i = tid; i < KPAD * EST; i += NTH) E[i] = 0.0f;
  if (tid < KPAD) pvec[tid] = 0.0f;

  // Prime the async pipeline with tiles t = 1 .. min(NBUF-1, len-1).
  const int pf_hi = (len - 1 < NBUF - 1) ? (len - 1) : (NBUF - 1);
  for (int pt = 1; pt <= pf_hi; ++pt) {
    const float* gsrc = sb + (size_t)pt * KKE;
    float* dst = raw[pt & (NBUF - 1)];
    for (int idx = tid; idx < KKE; idx += NTH)
      async_ld_b32((uint32_t)(size_t)&dst[idx], gsrc + idx);
  }

  // Gold score partial for this b (overlaps async-load latency; L2-warm later).
  float g = 0.0f;
  const int* tg = targets + (size_t)b * T;
  for (int t = tid; t < T; t += NTH)
    if (t < len) g += sb[(size_t)t * KKE + tg[t]];
  red[tid] = g;

  // alpha0 = scores[b, 0, START_TAG, :] (lengths >= 1 per reference; guard anyway)
  if (tid < KTAGS) alpha[tid] = (len > 0) ? sb[START_TAG * KTAGS + tid] : 0.0f;
  __syncthreads();
  for (int s = NTH >> 1; s > 0; s >>= 1) {
    if (tid < s) red[tid] += red[tid + s];
    __syncthreads();
  }
  const float gold = red[0];

  for (int t = 1; t < len; ++t) {
    // Wait until tile t has landed; tiles (t+1, t+2) may still be in flight.
    const int hi = (len - 1 < t + 2) ? (len - 1) : (t + 2);
    wait_async_groups(hi - t);
    __syncthreads();
    const float* rt = raw[t & (NBUF - 1)];

    // Keep the pipe full: issue tile t+3 into the buffer tile t-1 just vacated.
    if (t + 3 < len) {
      const float* gsrc = sb + (size_t)(t + 3) * KKE;
      float* dst = raw[(t + 3) & (NBUF - 1)];
      for (int idx = tid; idx < KKE; idx += NTH)
        async_ld_b32((uint32_t)(size_t)&dst[idx], gsrc + idx);
    }

    // Per-column max of the tile; max of alpha.
    if (tid < KTAGS) {
      float m = rt[tid];
      for (int i = 1; i < KTAGS; ++i) m = fmaxf(m, rt[i * KTAGS + tid]);
      cmax[tid] = m;
    }
    if (tid == 0) {
      float m = alpha[0];
      for (int i = 1; i < KTAGS; ++i) m = fmaxf(m, alpha[i]);
      amax_s = m;
    }
    __syncthreads();
    const float am = amax_s;
    if (tid < KTAGS) pvec[tid] = __expf(alpha[tid] - am);
    for (int idx = tid; idx < KKE; idx += NTH) {
      const int i = idx / KTAGS;
      const int j = idx - i * KTAGS;
      E[i * EST + j] = __expf(rt[idx] - cmax[j]);
    }
    __syncthreads();

    // v[j] = p . E[:,j] as a 1x64 * 64x64 matvec. Wave w owns N-tile w (16 cols),
    // chaining 16 K-chunks of V_WMMA_F32_16X16X4_F32. A rows replicated with p,
    // so D row 0 (== row 8) holds the result exactly in f32.
#if __has_builtin(__builtin_amdgcn_wmma_f32_16x16x4_f32)
    {
      const int wid  = tid >> 5;
      const int lane = tid & 31;
      const int n    = lane & 15;
      const int kb   = (lane & 16) ? 2 : 0;   // ISA 7.12.2: lanes 16-31 hold K=2,3
      v8f c = {};
      #pragma unroll
      for (int ch = 0; ch < 16; ++ch) {
        const int k0 = ch * 4 + kb;
        v2f a  = { pvec[k0], pvec[k0 + 1] };
        v2f bb = { E[k0 * EST + wid * 16 + n], E[(k0 + 1) * EST + wid * 16 + n] };
        c = __builtin_amdgcn_wmma_f32_16x16x4_f32(false, a, false, bb,
                                                  (short)0, c, false, false);
      }
      if (lane < 16) vout[wid * 16 + lane] = c[0];  // D row 0, N = wid*16 + lane
    }
#else
    if (tid < KTAGS) {
      float acc = 0.0f;
      for (int i = 0; i < KTAGS; ++i) acc += pvec[i] * E[i * EST + tid];
      vout[tid] = acc;
    }
#endif
    __syncthreads();
    if (tid < KTAGS) alpha[tid] = am + cmax[tid] + __logf(vout[tid]);
    __syncthreads();
  }

  if (tid == 0) partial[b] = alpha[END_TAG] - gold;
}

__global__ void crf_finalize_kernel(const float* __restrict__ partial,
                                    float* __restrict__ out, int B) {
  __shared__ float red[256];
  float s = 0.0f;
  for (int i = threadIdx.x; i < B; i += 256) s += partial[i];
  red[threadIdx.x] = s;
  __syncthreads();
  for (int k = 128; k > 0; k >>= 1) {
    if ((int)threadIdx.x < k) red[threadIdx.x] += red[threadIdx.x + k];
    __syncthreads();
  }
  if (threadIdx.x == 0) out[0] = red[0] / (float)B;
}

extern "C" void kernel_launch(void* const* d_in, const int* in_sizes, int n_in,
                              void* d_out, int out_size, void* d_ws, size_t ws_size,
                              hipStream_t stream) {
  const float* scores  = (const float*)d_in[0];
  const int*   targets = (const int*)d_in[1];
  const int*   lengths = (const int*)d_in[2];
  const int B = in_sizes[2];
  const int T = in_sizes[1] / B;          // K fixed at 50 (compile-time layout)
  float* partial = (float*)d_ws;          // B floats of scratch

  crf_fwd_kernel<<<B, NTH, 0, stream>>>(scores, targets, lengths, partial, T);
  crf_finalize_kernel<<<1, 256, 0, stream>>>(partial, (float*)d_out, B);
}